// CRFLoss_23527830847808
// MI455X (gfx1250) — compile-verified
//
#include <hip/hip_runtime.h>

typedef __attribute__((ext_vector_type(2))) float v2f;
typedef __attribute__((ext_vector_type(8))) float v8f;

#define B_ 128
#define T_ 1024
#define L_ 128
#define SMALL_ (-1000.0f)
#define MBLK 16

// ---------------------------------------------------------------------------
// Forward-algorithm scan. Each workgroup owns 16 batch rows. The logsumexp
// recurrence is rewritten as  new = m + log( exp(alpha-m) @ exp(trans) ) + emit
// so the per-step inner product is a 16x128x128 f32 GEMM done with
// V_WMMA_F32_16X16X4_F32. exp(trans) column tiles live in VGPRs (64/lane),
// exp(alpha - m) lives in LDS (8 KB), recurrence sync = 3 workgroup barriers.
// ---------------------------------------------------------------------------
__global__ __launch_bounds__(256) void crf_scan_kernel(
    const float* __restrict__ pred, const float* __restrict__ trans,
    const int* __restrict__ seq_len, float* __restrict__ lse_out)
{
    __shared__ float Aexp[MBLK][L_];     // exp(alpha - m) for current step
    __shared__ float mArr[2][MBLK];      // ping-pong row maxima
    __shared__ float partMax[8][MBLK];   // per-wave partial row max
    __shared__ float partSum[8][MBLK];   // per-wave partial row sum
    __shared__ int   slS[MBLK];

    const int tid   = threadIdx.x;
    const int wave  = tid >> 5;
    const int lane  = tid & 31;
    const int half  = lane >> 4;     // lanes 0-15 vs 16-31
    const int lcol  = lane & 15;
    const int rbase = half << 3;     // D tile: VGPR r <-> row r (+8 for upper half)
    const int koff  = half << 1;     // A/B 16x16x4: upper half holds K+2,K+3
    const int b0 = blockIdx.x * MBLK;
    const int n0 = wave * MBLK;
    const int START_ = L_ - 2, STOP_ = L_ - 1;

    // ---- one-time: B fragments = exp(transitions) column tile [128 x 16] ----
    v2f bf[32];
    #pragma unroll
    for (int kk = 0; kk < 32; ++kk) {
        int k0 = kk * 4 + koff;
        bf[kk].x = __expf(trans[k0 * L_ + n0 + lcol]);
        bf[kk].y = __expf(trans[(k0 + 1) * L_ + n0 + lcol]);
    }

    // ---- initial state: alpha0 = SMALL except START=0 -> Aexp one-hot, m=0 ----
    for (int i = tid; i < MBLK * L_; i += 256)
        (&Aexp[0][0])[i] = ((i & (L_ - 1)) == START_) ? 1.0f : 0.0f;
    if (tid < MBLK) { mArr[1][tid] = 0.0f; slS[tid] = seq_len[b0 + tid]; }
    __syncthreads();

    for (int t = 1; t <= T_ + 1; ++t) {
        const int cur = t & 1;

        // ---- S = exp(alpha-m) (16x128) @ expT (128x16), this wave's N tile ----
        v8f acc = {0.f, 0.f, 0.f, 0.f, 0.f, 0.f, 0.f, 0.f};
        #pragma unroll
        for (int kk = 0; kk < 32; ++kk) {
            v2f av = *reinterpret_cast<const v2f*>(&Aexp[lcol][kk * 4 + koff]);
            acc = __builtin_amdgcn_wmma_f32_16x16x4_f32(
                false, av, false, bf[kk], (short)0, acc, false, false);
        }

        // ---- new alpha values + emission (8 D elements per lane) ----
        float val[8];
        #pragma unroll
        for (int r = 0; r < 8; ++r) {
            const int grow = rbase + r;
            const int gcol = n0 + lcol;
            const int slp1 = slS[grow] + 1;
            float e;
            if (t == slp1) {
                e = (gcol == STOP_) ? 0.0f : SMALL_;           // forced STOP row
            } else if (t <= T_) {
                const size_t idx = ((size_t)(b0 + grow) * T_ + (t - 1)) * L_ + gcol;
                e = pred[idx];
                if (gcol >= START_) e = SMALL_;                // mask START/STOP cols
                if (t < T_) __builtin_prefetch(&pred[idx + L_], 0, 1);
            } else {
                e = 0.0f;                                      // end_state zeros
            }
            float s = fmaxf(acc[r], 1e-37f);
            val[r] = mArr[cur][grow] + __logf(s) + e;
        }

        // ---- row max across 16 cols: shuffle within 16-lane halves ----
        #pragma unroll
        for (int r = 0; r < 8; ++r) {
            float m = val[r];
            m = fmaxf(m, __shfl_xor(m, 1, 32));
            m = fmaxf(m, __shfl_xor(m, 2, 32));
            m = fmaxf(m, __shfl_xor(m, 4, 32));
            m = fmaxf(m, __shfl_xor(m, 8, 32));
            if (lcol == 0) partMax[wave][rbase + r] = m;
        }
        __syncthreads();   // B1: Aexp reads done, partMax ready

        float mf = 0.0f;
        if (wave == 0 && lane < MBLK) {
            mf = partMax[0][lane];
            #pragma unroll
            for (int w = 1; w < 8; ++w) mf = fmaxf(mf, partMax[w][lane]);
            mArr[cur ^ 1][lane] = mf;
        }
        __syncthreads();   // B2: final row maxima visible

        // ---- write exp(new - m2) into Aexp for next step, partial row sums ----
        #pragma unroll
        for (int r = 0; r < 8; ++r) {
            const int grow = rbase + r;
            float ex = __expf(val[r] - mArr[cur ^ 1][grow]);
            Aexp[grow][n0 + lcol] = ex;
            ex += __shfl_xor(ex, 1, 32);
            ex += __shfl_xor(ex, 2, 32);
            ex += __shfl_xor(ex, 4, 32);
            ex += __shfl_xor(ex, 8, 32);
            if (lcol == 0) partSum[wave][grow] = ex;
        }
        __syncthreads();   // B3: Aexp ready for next step, sums ready

        // ---- capture logsumexp of row b at its terminal step (t == sl+1) ----
        if (wave == 0 && lane < MBLK && (slS[lane] + 1) == t) {
            float s = partSum[0][lane];
            #pragma unroll
            for (int w = 1; w < 8; ++w) s += partSum[w][lane];
            lse_out[b0 + lane] = mf + __logf(s);
        }
    }
}

// ---------------------------------------------------------------------------
// Gold-path score: emissions along the tag path + transition chain with
// START/STOP padding, one block per batch row.
// ---------------------------------------------------------------------------
__global__ __launch_bounds__(256) void crf_path_kernel(
    const float* __restrict__ pred, const float* __restrict__ trans,
    const int* __restrict__ tags, const int* __restrict__ seq_len,
    float* __restrict__ path_out)
{
    __shared__ float red[256];
    const int b  = blockIdx.x;
    const int sl = seq_len[b];
    const int START_ = L_ - 2, STOP_ = L_ - 1;
    float s = 0.0f;
    for (int t = threadIdx.x; t < sl; t += 256)
        s += pred[((size_t)b * T_ + t) * L_ + tags[b * T_ + t]];
    for (int k = threadIdx.x; k <= sl; k += 256) {
        const int from = (k == 0)  ? START_ : tags[b * T_ + k - 1];
        const int to   = (k == sl) ? STOP_  : tags[b * T_ + k];
        s += trans[from * L_ + to];
    }
    red[threadIdx.x] = s;
    __syncthreads();
    for (int off = 128; off > 0; off >>= 1) {
        if (threadIdx.x < off) red[threadIdx.x] += red[threadIdx.x + off];
        __syncthreads();
    }
    if (threadIdx.x == 0) path_out[b] = red[0];
}

// ---------------------------------------------------------------------------
// Final scalar: sum_b lse[b] - sum_b path[b]  (deterministic tree reduce)
// ---------------------------------------------------------------------------
__global__ __launch_bounds__(128) void crf_finish_kernel(
    const float* __restrict__ lse, const float* __restrict__ path,
    float* __restrict__ out)
{
    __shared__ float red[128];
    red[threadIdx.x] = lse[threadIdx.x] - path[threadIdx.x];
    __syncthreads();
    for (int off = 64; off > 0; off >>= 1) {
        if (threadIdx.x < off) red[threadIdx.x] += red[threadIdx.x + off];
        __syncthreads();
    }
    if (threadIdx.x == 0) out[0] = red[0];
}

extern "C" void kernel_launch(void* const* d_in, const int* in_sizes, int n_in,
                              void* d_out, int out_size, void* d_ws, size_t ws_size,
                              hipStream_t stream) {
    const float* pred    = (const float*)d_in[0];
    const float* trans   = (const float*)d_in[1];
    const int*   tags    = (const int*)d_in[2];
    const int*   seq_len = (const int*)d_in[3];
    float* lse  = (float*)d_ws;        // 128 floats
    float* path = lse + B_;            // 128 floats

    crf_scan_kernel<<<B_ / MBLK, 256, 0, stream>>>(pred, trans, seq_len, lse);
    crf_path_kernel<<<B_, 256, 0, stream>>>(pred, trans, tags, seq_len, path);
    crf_finish_kernel<<<1, B_, 0, stream>>>(lse, path, (float*)d_out);
}